// Attention_36009005809852
// MI455X (gfx1250) — compile-verified
//
#include <hip/hip_runtime.h>
#include <cstddef>

// Problem constants (match reference)
#define NB   2048   // batches B
#define PPOS 128    // positions P
#define DDIM 64     // D
#define NH   8      // heads H
#define DB   10     // b feature dim
#define HD   512    // H*D
#define KV2  1024   // 2*H*D

// LDS row strides (padded to dodge 64-bank conflicts)
#define XS 68       // x tile row stride (64 cols)
#define QS 516      // q / vals tile row stride (512 cols)
#define KS 1028     // kvx tile row stride (1024 cols)
#define AS 9        // attn row stride (8 cols)

// float offsets into dynamic LDS
#define OFF_X   0
#define OFF_Q   (OFF_X  + 16*XS)        // 1088
#define OFF_KV  (OFF_Q  + 16*QS)        // 9344
#define OFF_G   (OFF_KV + 16*KS)        // 25792  [16][10][8]
#define OFF_C0  (OFF_G  + 16*DB*NH)     // 27072  [16][8]
#define OFF_B   (OFF_C0 + 16*NH)        // 27200  [128][10]
#define OFF_A   (OFF_B  + PPOS*DB)      // 28480  [128][9]
#define OFF_T   (OFF_A  + PPOS*AS)      // 29632  [8][10]
#define OFF_S   (OFF_T  + NH*DB)        // 29712  [8]
#define LDS_FLOATS (OFF_S + NH)         // 29720 -> 118880 bytes

typedef __attribute__((ext_vector_type(2))) float v2f;
typedef __attribute__((ext_vector_type(8))) float v8f;

__device__ __forceinline__ v8f wmma_f32_16x16x4(v2f a, v2f b, v8f c) {
    // D = A(16x4) * B(4x16) + C(16x16), fp32, wave32
    return __builtin_amdgcn_wmma_f32_16x16x4_f32(
        /*neg_a=*/false, a, /*neg_b=*/false, b,
        /*c_mod=*/(short)0, c, /*reuse_a=*/false, /*reuse_b=*/false);
}

extern "C" __global__ __launch_bounds__(128)
void attn_fused_kernel(const float* __restrict__ x,
                       const float* __restrict__ bmat,
                       const float* __restrict__ Wkvx,
                       const float* __restrict__ bkvx,
                       const float* __restrict__ Wkvb,
                       const float* __restrict__ bkvb,
                       const float* __restrict__ Wq,
                       const float* __restrict__ bq,
                       const float* __restrict__ Wf,
                       const float* __restrict__ bf,
                       float* __restrict__ out)
{
    extern __shared__ float smem[];
    float* xs  = smem + OFF_X;
    float* qsv = smem + OFF_Q;    // q tile, later reused as vals tile
    float* kvs = smem + OFF_KV;   // x @ W_kvx tile (no bias yet)
    float* gsm = smem + OFF_G;    // G[bt][e][h]
    float* c0s = smem + OFF_C0;   // c0[bt][h]
    float* bls = smem + OFF_B;    // b rows for current bt
    float* als = smem + OFF_A;    // attn rows for current bt
    float* Tsm = smem + OFF_T;    // T[h][e]
    float* Ssm = smem + OFF_S;    // S[h]

    const int t     = threadIdx.x;
    const int lane  = t & 31;
    const int wave  = t >> 5;       // 4 waves
    const int lmod  = lane & 15;    // fragment row/col index
    const int lhi   = lane >> 4;    // lane half
    const int khi   = lhi * 2;      // K offset within fragment
    const int btile = blockIdx.x * 16;

    // ---- Stage 1: load x tile [16 x 64] (contiguous in global) ----
    for (int i = t; i < 16 * DDIM; i += 128) {
        xs[(i >> 6) * XS + (i & 63)] = x[(size_t)btile * DDIM + i];
    }
    __syncthreads();

    // ---- Stage 2: q = x @ Wq + bq   (M=16, N=512, K=64, WMMA) ----
    for (int nt = wave; nt < HD / 16; nt += 4) {
        const int ncol = nt * 16 + lmod;
        v8f acc = {};
        for (int k0 = 0; k0 < DDIM; k0 += 4) {
            v2f a, bfrag;
            a.x     = xs[lmod * XS + k0 + khi];
            a.y     = xs[lmod * XS + k0 + khi + 1];
            bfrag.x = Wq[(size_t)(k0 + khi)     * HD + ncol];
            bfrag.y = Wq[(size_t)(k0 + khi + 1) * HD + ncol];
            acc = wmma_f32_16x16x4(a, bfrag, acc);
        }
        const float bias = bq[ncol];
#pragma unroll
        for (int j = 0; j < 8; ++j)
            qsv[(j + 8 * lhi) * QS + ncol] = acc[j] + bias;
    }

    // ---- Stage 3: kvx = x @ Wkvx    (M=16, N=1024, K=64, WMMA) ----
    for (int nt = wave; nt < KV2 / 16; nt += 4) {
        const int ncol = nt * 16 + lmod;
        v8f acc = {};
        for (int k0 = 0; k0 < DDIM; k0 += 4) {
            v2f a, bfrag;
            a.x     = xs[lmod * XS + k0 + khi];
            a.y     = xs[lmod * XS + k0 + khi + 1];
            bfrag.x = Wkvx[(size_t)(k0 + khi)     * KV2 + ncol];
            bfrag.y = Wkvx[(size_t)(k0 + khi + 1) * KV2 + ncol];
            acc = wmma_f32_16x16x4(a, bfrag, acc);
        }
#pragma unroll
        for (int j = 0; j < 8; ++j)
            kvs[(j + 8 * lhi) * KS + ncol] = acc[j];
    }
    __syncthreads();

    // ---- Stage 4: per (bt,h): c0 and G[e][h] ----
    {
        const int bt = t >> 3;   // 0..15
        const int h  = t & 7;    // 0..7
        const float* qrow = qsv  + bt * QS + h * DDIM;
        const float* krow = kvs  + bt * KS + h * DDIM;   // k half
        const float* bk1  = bkvx + h * DDIM;
        const float* bk2  = bkvb + h * DDIM;
        float c = 0.0f;
        float gacc[DB];
#pragma unroll
        for (int e = 0; e < DB; ++e) gacc[e] = 0.0f;
        for (int d = 0; d < DDIM; ++d) {
            const float qv = qrow[d];
            c += (krow[d] + bk1[d] + bk2[d]) * qv;
#pragma unroll
            for (int e = 0; e < DB; ++e)
                gacc[e] += Wkvb[(size_t)e * KV2 + h * DDIM + d] * qv;
        }
        c0s[bt * NH + h] = c;
#pragma unroll
        for (int e = 0; e < DB; ++e)
            gsm[(bt * DB + e) * NH + h] = gacc[e];
    }
    __syncthreads();

    // ---- Stage 5: per-batch attention via rank-10 structure ----
    const float scale = 0.125f;  // 1/sqrt(64)
    for (int bt = 0; bt < 16; ++bt) {
        // load b rows for this batch: [128 x 10], contiguous
        const float* bsrc = bmat + (size_t)(btile + bt) * PPOS * DB;
        for (int i = t; i < PPOS * DB; i += 128) bls[i] = bsrc[i];
        __syncthreads();

        // one thread per position p: dots + softmax over heads
        {
            const int p = t;
            float br[DB];
#pragma unroll
            for (int e = 0; e < DB; ++e) br[e] = bls[p * DB + e];
            float dt[NH];
#pragma unroll
            for (int h = 0; h < NH; ++h) dt[h] = c0s[bt * NH + h];
#pragma unroll
            for (int e = 0; e < DB; ++e) {
                const float be = br[e];
#pragma unroll
                for (int h = 0; h < NH; ++h)
                    dt[h] += be * gsm[(bt * DB + e) * NH + h];
            }
            float mx = dt[0] * scale;
#pragma unroll
            for (int h = 1; h < NH; ++h) mx = fmaxf(mx, dt[h] * scale);
            float ex[NH], s = 0.0f;
#pragma unroll
            for (int h = 0; h < NH; ++h) { ex[h] = __expf(dt[h] * scale - mx); s += ex[h]; }
            const float inv = 1.0f / s;
#pragma unroll
            for (int h = 0; h < NH; ++h) als[p * AS + h] = ex[h] * inv;
        }
        __syncthreads();

        // reductions over P: T[h][e] and S[h]
        if (t < NH * DB) {
            const int h = t / DB, e = t % DB;
            float acc = 0.0f;
            for (int p = 0; p < PPOS; ++p) acc += als[p * AS + h] * bls[p * DB + e];
            Tsm[h * DB + e] = acc;
        } else if (t < NH * DB + NH) {
            const int h = t - NH * DB;
            float acc = 0.0f;
            for (int p = 0; p < PPOS; ++p) acc += als[p * AS + h];
            Ssm[h] = acc;
        }
        __syncthreads();

        // vals[bt][f] = S[h]*vconst + sum_e T[h][e]*Wv[e][f]  (4 per thread)
#pragma unroll
        for (int i2 = 0; i2 < 4; ++i2) {
            const int f = t * 4 + i2;       // 0..511
            const int h = f >> 6;
            const float vconst = kvs[bt * KS + HD + f] + bkvx[HD + f] + bkvb[HD + f];
            float val = Ssm[h] * vconst;
#pragma unroll
            for (int e = 0; e < DB; ++e)
                val += Tsm[h * DB + e] * Wkvb[(size_t)e * KV2 + HD + f];
            qsv[bt * QS + f] = val;         // q tile reused as vals tile
        }
        __syncthreads();
    }

    // ---- Stage 6: out = vals @ Wf + bf  (M=16, N=64, K=512, WMMA) ----
    {
        const int nt   = wave;              // 4 waves cover N=64
        const int ncol = nt * 16 + lmod;
        v8f acc = {};
        for (int k0 = 0; k0 < HD; k0 += 4) {
            v2f a, bfrag;
            a.x     = qsv[lmod * QS + k0 + khi];
            a.y     = qsv[lmod * QS + k0 + khi + 1];
            bfrag.x = Wf[(size_t)(k0 + khi)     * DDIM + ncol];
            bfrag.y = Wf[(size_t)(k0 + khi + 1) * DDIM + ncol];
            acc = wmma_f32_16x16x4(a, bfrag, acc);
        }
        const float bias = bf[ncol];
#pragma unroll
        for (int j = 0; j < 8; ++j) {
            const int m = j + 8 * lhi;
            out[(size_t)(btile + m) * DDIM + ncol] = acc[j] + bias;
        }
    }
}

extern "C" void kernel_launch(void* const* d_in, const int* in_sizes, int n_in,
                              void* d_out, int out_size, void* d_ws, size_t ws_size,
                              hipStream_t stream) {
    (void)in_sizes; (void)n_in; (void)out_size; (void)d_ws; (void)ws_size;
    const float* x    = (const float*)d_in[0];
    const float* bmat = (const float*)d_in[1];
    const float* Wkvx = (const float*)d_in[2];
    const float* bkvx = (const float*)d_in[3];
    const float* Wkvb = (const float*)d_in[4];
    const float* bkvb = (const float*)d_in[5];
    const float* Wq   = (const float*)d_in[6];
    const float* bq   = (const float*)d_in[7];
    const float* Wf   = (const float*)d_in[8];
    const float* bf   = (const float*)d_in[9];
    float* out = (float*)d_out;

    const size_t smem_bytes = (size_t)LDS_FLOATS * sizeof(float); // ~116 KB (<320 KB/WG on gfx1250)
    dim3 grid(NB / 16), block(128);
    attn_fused_kernel<<<grid, block, smem_bytes, stream>>>(
        x, bmat, Wkvx, bkvx, Wkvb, bkvb, Wq, bq, Wf, bf, out);
}